// MNN_70497593197435
// MI455X (gfx1250) — compile-verified
//
#include <hip/hip_runtime.h>
#include <stdint.h>

#define B_ 8
#define C_ 3
#define H_ 128
#define W_ 128
#define O_ 8
#define KS_ 5
#define P_ (KS_ * KS_)
#define HO_ 124
#define WO_ 124
#define TILE_H 4
#define IN_ROWS (TILE_H + KS_ - 1) /* 8 */
#define NT (HO_ / TILE_H)          /* 31 */
#define NPIX (TILE_H * WO_)        /* 496 */

typedef uint32_t v4u __attribute__((ext_vector_type(4)));
typedef int32_t  v8i __attribute__((ext_vector_type(8)));
typedef int32_t  v4i __attribute__((ext_vector_type(4)));

__global__ __launch_bounds__(256) void hitmiss_tdm_kernel(
    const float* __restrict__ x,
    const float* __restrict__ Kh,
    const float* __restrict__ Km,
    float* __restrict__ out)
{
    __shared__ float sx[C_][IN_ROWS][W_];   // 12 KB tile: 3 ch x 8 rows x 128 cols

    const int blk  = blockIdx.x;
    const int b    = blk / NT;
    const int tile = blk - b * NT;
    const int r0   = tile * TILE_H;         // first output row of this tile

    // ---- TDM: DMA x[b][0:3][r0:r0+8][0:128] -> LDS (wave 0 issues) ----
    if (threadIdx.x < 32) {
        const uint64_t gaddr =
            (uint64_t)(uintptr_t)(x + ((size_t)b * C_ * H_ + (size_t)r0) * W_);
        const uint32_t laddr = (uint32_t)(uintptr_t)(&sx[0][0][0]);

        v4u g0;
        g0[0] = 1u;                                   // count=1, user descriptor
        g0[1] = laddr;                                // lds_addr (bytes)
        g0[2] = (uint32_t)(gaddr & 0xFFFFFFFFu);      // global_addr[31:0]
        g0[3] = (uint32_t)((gaddr >> 32) & 0x1FFFFFFu) | (2u << 30); // [56:32]|type=2

        v8i g1;
        g1[0] = (int)(2u << 16);                      // wg_mask=0, data_size=2 (4B)
        g1[1] = (int)(((uint32_t)W_) << 16);          // tensor_dim0 lo16 = 128
        g1[2] = (int)(((uint32_t)H_) << 16);          // td0 hi=0 | tensor_dim1 lo16 = 128
        g1[3] = (int)(((uint32_t)W_) << 16);          // td1 hi=0 | tile_dim0 = 128
        g1[4] = (int)((uint32_t)IN_ROWS |             // tile_dim1 = 8
                      (((uint32_t)C_) << 16));        // tile_dim2 = 3
        g1[5] = W_;                                   // tensor_dim0_stride = 128
        g1[6] = (int)(((uint32_t)(H_ * W_) & 0xFFFFu) << 16); // td1_stride lo16=16384
        g1[7] = (int)((uint32_t)(H_ * W_) >> 16);     // td1_stride[47:16] = 0

        v4i g2;
        g2[0] = C_;                                   // tensor_dim2 = 3
        g2[1] = 0; g2[2] = 0; g2[3] = 0;              // no dim3, no iterate

        v4i g3;
        g3[0] = 0; g3[1] = 0; g3[2] = 0; g3[3] = 0;

        v8i g4;                                       // extra group (clang-23 form)
        g4[0] = 0; g4[1] = 0; g4[2] = 0; g4[3] = 0;
        g4[4] = 0; g4[5] = 0; g4[6] = 0; g4[7] = 0;

        __builtin_amdgcn_tensor_load_to_lds(g0, g1, g2, g3, g4, 0);
        __builtin_amdgcn_s_wait_tensorcnt(0);
    }
    __syncthreads();

    // ---- compute: each thread owns up to 2 pixels of the 4x124 tile ----
    const int  t     = threadIdx.x;
    const int  idx0  = t;                 // always < 496
    const int  idx1  = t + 256;
    const bool hasB  = (idx1 < NPIX);
    const int  idx1c = hasB ? idx1 : 0;

    const int rA = idx0 / WO_, cA = idx0 - rA * WO_;
    const int rB = idx1c / WO_, cB = idx1c - rB * WO_;

    float pa[P_], pb[P_];

    for (int c = 0; c < C_; ++c) {
        // gather both 5x5 patches from LDS (reused across all 8 output maps)
        #pragma unroll
        for (int u = 0; u < KS_; ++u) {
            #pragma unroll
            for (int v = 0; v < KS_; ++v) {
                pa[u * KS_ + v] = sx[c][rA + u][cA + v];
                pb[u * KS_ + v] = sx[c][rB + u][cB + v];
            }
        }

        for (int o = 0; o < O_; ++o) {
            // uniform addresses -> scalar loads; K values land in SGPRs and
            // feed v_sub_f32 directly as scalar operands.
            const float* __restrict__ kh = Kh + (size_t)(o * C_ + c) * P_;
            const float* __restrict__ km = Km + (size_t)(o * C_ + c) * P_;

            float hitA  = pa[0] - kh[0];
            float missA = pa[0] - km[0];
            float hitB  = pb[0] - kh[0];
            float missB = pb[0] - km[0];
            #pragma unroll
            for (int p = 1; p < P_; ++p) {
                const float khp = kh[p];
                const float kmp = km[p];
                hitA  = fminf(hitA,  pa[p] - khp);   // fuses to v_min3_num_f32
                missA = fmaxf(missA, pa[p] - kmp);   // fuses to v_max3_num_f32
                hitB  = fminf(hitB,  pb[p] - khp);
                missB = fmaxf(missB, pb[p] - kmp);
            }

            const size_t ch = (size_t)b * (O_ * C_) + (size_t)(o * C_ + c);
            out[(ch * HO_ + (size_t)(r0 + rA)) * WO_ + cA] = hitA - missA;
            if (hasB)
                out[(ch * HO_ + (size_t)(r0 + rB)) * WO_ + cB] = hitB - missB;
        }
    }
}

extern "C" void kernel_launch(void* const* d_in, const int* in_sizes, int n_in,
                              void* d_out, int out_size, void* d_ws, size_t ws_size,
                              hipStream_t stream)
{
    (void)in_sizes; (void)n_in; (void)out_size; (void)d_ws; (void)ws_size;
    const float* x  = (const float*)d_in[0];
    const float* Kh = (const float*)d_in[1];
    const float* Km = (const float*)d_in[2];
    float*       o  = (float*)d_out;

    hipLaunchKernelGGL(hitmiss_tdm_kernel,
                       dim3(B_ * NT), dim3(256), 0, stream,
                       x, Kh, Km, o);
}